// SSKernelDiag_2619930050871
// MI455X (gfx1250) — compile-verified
//
#include <hip/hip_runtime.h>

// S4D / SSKernelDiag log-Vandermonde kernel for gfx1250 (MI455X).
// K[h,l] = 2*Re( sum_n (C*B)[h,n] * dA[h,n]^l ),  dA = (1+dtA/2)/(1-dtA/2)
// Mapped onto V_WMMA_F32_16X16X4_F32: per (h, l-tile of 256), a real
// 16x128 @ 128x16 GEMM (re/im interleaved along K) = 32 chained K=4 WMMAs.

typedef float v2f __attribute__((ext_vector_type(2)));
typedef float v8f __attribute__((ext_vector_type(8)));

#define H_DIM 1024
#define N_DIM 64
#define L_DIM 1024

struct c2 { float re, im; };

__device__ __forceinline__ c2 cmul(c2 a, c2 b) {
    return c2{ fmaf(a.re, b.re, -(a.im * b.im)),
               fmaf(a.re, b.im,  (a.im * b.re)) };
}
__device__ __forceinline__ c2 csqr(c2 a) {
    return c2{ fmaf(a.re, a.re, -(a.im * a.im)), 2.0f * a.re * a.im };
}

__global__ __launch_bounds__(128)
void s4d_vandermonde_wmma(const float* __restrict__ log_dt,
                          const float* __restrict__ inv_A_real,
                          const float* __restrict__ A_imag,
                          const float* __restrict__ B_ri,
                          const float* __restrict__ C_ri,
                          float* __restrict__ out)
{
    // Per-h LDS tables (one block == one h; 18 KB total)
    __shared__ float sB[N_DIM][16][2];   // (Re dA^j, -Im dA^j),  j = 0..15
    __shared__ float sA[N_DIM][16][2];   // Cterm * dA^(16m),     m = 0..15
    __shared__ float sF[4][N_DIM][2];    // dA^(256*tile),        tile = 0..3

    const int h   = blockIdx.x;
    const int tid = threadIdx.x;

    // ---------------- setup: bilinear discretization + power tables --------
    {
        const int n   = (tid < N_DIM) ? tid : (tid - N_DIM);
        const int idx = h * N_DIM + n;

        const float dt = __expf(log_dt[h]);
        const float ar = -__expf(inv_A_real[idx]);
        const float ai = A_imag[idx];
        // x = dt*A/2 ; dA = (1+x)/(1-x)
        const float xr  = 0.5f * dt * ar;
        const float xi  = 0.5f * dt * ai;
        const float den = (1.0f - xr) * (1.0f - xr) + xi * xi;
        const float rdn = 1.0f / den;
        const c2 dA = { (1.0f - xr * xr - xi * xi) * rdn, (2.0f * xi) * rdn };

        if (tid < N_DIM) {
            // Cterm = C * B (complex)
            const float br = B_ri[2 * idx], bi = B_ri[2 * idx + 1];
            const float cr = C_ri[2 * idx], ci = C_ri[2 * idx + 1];
            const c2 ct = { fmaf(cr, br, -(ci * bi)), fmaf(cr, bi, ci * br) };

            // B table: conj(dA^j), j = 0..15
            c2 v = { 1.0f, 0.0f };
            #pragma unroll
            for (int j = 0; j < 16; ++j) {
                sB[n][j][0] = v.re;
                sB[n][j][1] = -v.im;
                v = cmul(v, dA);
            }
            // now v = dA^16 ; A table: Cterm * dA^(16m), m = 0..15
            c2 a = ct;
            #pragma unroll
            for (int m = 0; m < 16; ++m) {
                sA[n][m][0] = a.re;
                sA[n][m][1] = a.im;
                a = cmul(a, v);
            }
        } else {
            // dA^256 via 8 squarings, then tile factors dA^(256*t), t = 0..3
            c2 r = dA;
            #pragma unroll
            for (int k = 0; k < 8; ++k) r = csqr(r);
            c2 f = { 1.0f, 0.0f };
            #pragma unroll
            for (int t = 0; t < 4; ++t) {
                sF[t][n][0] = f.re;
                sF[t][n][1] = f.im;
                f = cmul(f, r);
            }
        }
    }
    __syncthreads();

    // ---------------- main loop: 32 chained f32 WMMAs per wave -------------
    const int lane = tid & 31;
    const int tile = tid >> 5;     // wave id == l-tile (0..3)
    const int q    = lane & 15;    // A-row m and B-col j for this lane
    const int p    = lane >> 4;    // mode parity: n = 2s + p

    v8f acc = { 0.f, 0.f, 0.f, 0.f, 0.f, 0.f, 0.f, 0.f };

    #pragma unroll
    for (int s = 0; s < 32; ++s) {
        const int n = 2 * s + p;
        const v2f af = *(const v2f*)&sA[n][q][0];      // Cterm*dA^(16q)
        const v2f ff = *(const v2f*)&sF[tile][n][0];   // dA^(256*tile)
        const v2f bf = *(const v2f*)&sB[n][q][0];      // conj(dA^q)

        const c2 a = cmul(c2{ af.x, af.y }, c2{ ff.x, ff.y });
        const v2f A = { a.re, a.im };

        // D = A(16x4) * B(4x16) + C  — K-slice k = 4s..4s+3 of the 128-deep
        // re/im-interleaved contraction; accumulate across all 32 steps.
        acc = __builtin_amdgcn_wmma_f32_16x16x4_f32(
                  false, A, false, bf, (short)0, acc, false, false);
    }

    // D layout: VGPR r, lanes 0-15 -> M=r, lanes 16-31 -> M=r+8, N=lane%16
    // l = 256*tile + 16*(r + 8p) + q
    float* op = out + h * L_DIM + tile * 256 + p * 128 + q;
    #pragma unroll
    for (int r = 0; r < 8; ++r)
        op[16 * r] = 2.0f * acc[r];
}

extern "C" void kernel_launch(void* const* d_in, const int* in_sizes, int n_in,
                              void* d_out, int out_size, void* d_ws, size_t ws_size,
                              hipStream_t stream) {
    (void)in_sizes; (void)n_in; (void)out_size; (void)d_ws; (void)ws_size;
    const float* log_dt     = (const float*)d_in[0];
    const float* inv_A_real = (const float*)d_in[1];
    const float* A_imag     = (const float*)d_in[2];
    const float* B_ri       = (const float*)d_in[3];
    const float* C_ri       = (const float*)d_in[4];
    // d_in[5] is L (int scalar) == 1024, compiled in as L_DIM.
    float* out = (float*)d_out;

    s4d_vandermonde_wmma<<<dim3(H_DIM), dim3(128), 0, stream>>>(
        log_dt, inv_A_real, A_imag, B_ri, C_ri, out);
}